// ODEFunc_88897233092948
// MI455X (gfx1250) — compile-verified
//
#include <hip/hip_runtime.h>
#include <hip/hip_bf16.h>
#include <math.h>
#include <stdint.h>

typedef __attribute__((ext_vector_type(2))) float v2f;
typedef __attribute__((ext_vector_type(8))) float v8f;

#define NN    4096
#define BB    32
#define LAT   16
#define UNITS 64
#define NNZE  65536
#define MM    5   // Chebyshev terms: x0, s1x1, s1x2, s2x1, s2x2

// ---------------------------------------------------------------------------
// y (B, N*LAT) -> x0 (N, LAT*B):  x0[n, f*B + b] = y[b, n*LAT + f]
// ---------------------------------------------------------------------------
__global__ void permute_y_kernel(const float* __restrict__ y,
                                 float* __restrict__ x0) {
    int idx = blockIdx.x * blockDim.x + threadIdx.x;   // over N*LAT*B
    if (idx >= NN * LAT * BB) return;
    int b = idx & 31;
    int f = (idx >> 5) & 15;
    int n = idx >> 9;
    x0[idx] = y[(size_t)b * (NN * LAT) + n * LAT + f];
}

__global__ void fill_zero_kernel(float4* __restrict__ p, int n4) {
    int i = blockIdx.x * blockDim.x + threadIdx.x;
    if (i < n4) p[i] = make_float4(0.f, 0.f, 0.f, 0.f);
}

// dst = -src  (Chebyshev x2 = 2*S*x1 - x0 : start from -x0, atomic-add 2*S*x1)
__global__ void neg_copy_kernel(float4* __restrict__ dst,
                                const float4* __restrict__ src, int n4) {
    int i = blockIdx.x * blockDim.x + threadIdx.x;
    if (i < n4) {
        float4 v = src[i];
        dst[i] = make_float4(-v.x, -v.y, -v.z, -v.w);
    }
}

// acc (N, out, B) initialized to broadcast bias[o]
__global__ void bias_init_kernel(float* __restrict__ acc,
                                 const float* __restrict__ bias, int out) {
    int idx = blockIdx.x * blockDim.x + threadIdx.x;   // over N*out*B
    if (idx >= NN * out * BB) return;
    int o = (idx >> 5) % out;
    acc[idx] = bias[o];
}

// ---------------------------------------------------------------------------
// SpMM scatter: out[rows[e], j] += scale * vals[e] * x[cols[e], j]
// One thread handles 4 contiguous columns (float4 gather, 4 f32 atomics).
// ---------------------------------------------------------------------------
__global__ void spmm_atomic_kernel(const int*   __restrict__ rows,
                                   const int*   __restrict__ cols,
                                   const float* __restrict__ vals,
                                   const float* __restrict__ x,
                                   float*       __restrict__ out,
                                   int FB, float scale) {
    int tid = blockIdx.x * blockDim.x + threadIdx.x;
    int per_edge = FB >> 2;
    int e = tid / per_edge;
    if (e >= NNZE) return;
    int j = (tid % per_edge) << 2;
    int r = rows[e];
    int c = cols[e];
    float v = vals[e] * scale;
    float4 xv = *(const float4*)(x + (size_t)c * FB + j);
    float* o = out + (size_t)r * FB + j;
    atomicAdd(o + 0, v * xv.x);
    atomicAdd(o + 1, v * xv.y);
    atomicAdd(o + 2, v * xv.z);
    atomicAdd(o + 3, v * xv.w);
}

// ---------------------------------------------------------------------------
// WMMA GEMM-accumulate for one diffusion term m:
//   acc[n, o, b] += sum_f X[n, f*B + b] * W[f*MM + m, o]
//
// Block = 256 threads = 8 waves; NPB = 8 / ((OUT/16)*2) nodes per block.
// X tile staged into LDS with gfx1250 async-to-LDS (ASYNCcnt); W m-slice
// staged transposed (sW[o][k]) so A-fragments are contiguous ds_load_b64.
// One wave per 16x16 tile; V_WMMA_F32_16X16X4_F32, unrolled K-loop.
// Fragment layouts per ISA 7.12.2.
// ---------------------------------------------------------------------------
template <int F, int OUT>
__global__ void __launch_bounds__(256)
gemm_wmma_lds_kernel(const float* __restrict__ X,    // (N, F*B)
                     const float* __restrict__ W,    // (F*MM, OUT)
                     float*       __restrict__ acc,  // (N, OUT, B)
                     int m_idx) {
    constexpr int TPN = (OUT / 16) * 2;   // tiles (waves) per node
    constexpr int NPB = 8 / TPN;          // nodes per block
    constexpr int XT  = NPB * F * BB;     // staged X floats

    __shared__ float sX[XT];
    __shared__ float sW[F * OUT];

    const int tid = threadIdx.x;
    const int n0  = blockIdx.x * NPB;

    // ---- stage X tile: global -> LDS via async copy (ASYNCcnt) ------------
    {
        uint64_t gb = (uint64_t)(uintptr_t)(X + (size_t)n0 * F * BB);
        uint32_t lb = (uint32_t)(uintptr_t)sX;   // flat LDS addr[31:0] = DS offset
        constexpr int EPT = XT / 256;            // floats per thread
        if constexpr ((EPT & 3) == 0) {
#pragma unroll
            for (int i = 0; i < EPT / 4; ++i) {
                uint32_t off = (uint32_t)(tid + i * 256) * 16u;
                uint32_t lds = lb + off;
                uint64_t ga  = gb + off;
                asm volatile("global_load_async_to_lds_b128 %0, %1, off"
                             :: "v"(lds), "v"(ga) : "memory");
            }
        } else {
#pragma unroll
            for (int i = 0; i < EPT; ++i) {
                uint32_t off = (uint32_t)(tid + i * 256) * 4u;
                uint32_t lds = lb + off;
                uint64_t ga  = gb + off;
                asm volatile("global_load_async_to_lds_b32 %0, %1, off"
                             :: "v"(lds), "v"(ga) : "memory");
            }
        }
    }

    // ---- stage W m-slice transposed: sW[o*F + k] = W[(k*MM+m)*OUT + o] ----
#pragma unroll
    for (int i = tid; i < F * OUT; i += 256) {
        int o = i / F;
        int k = i % F;
        sW[i] = W[(size_t)(k * MM + m_idx) * OUT + o];
    }

    asm volatile("s_wait_asynccnt 0x0" ::: "memory");
    __syncthreads();

    // ---- per-wave 16x16 tile ----------------------------------------------
    const int lane = tid & 31;
    const int w    = tid >> 5;
    const int nl   = w / TPN;              // node within block
    const int t    = w % TPN;
    const int o0   = (t >> 1) << 4;
    const int b0   = (t & 1) << 4;
    const int half = lane >> 4;            // 0: lanes 0-15, 1: lanes 16-31
    const int lid  = lane & 15;

    float* accbase = acc + (size_t)(n0 + nl) * OUT * BB;
    v8f c;
#pragma unroll
    for (int r = 0; r < 8; ++r)
        c[r] = accbase[(o0 + r + half * 8) * BB + b0 + lid];

    const float* sXn = sX + nl * F * BB;
    const float* sWo = sW + (o0 + lid) * F;
#pragma unroll
    for (int k0 = 0; k0 < F; k0 += 4) {
        const int ka = k0 + half * 2;
        v2f a = *(const v2f*)(sWo + ka);           // ds_load_b64
        v2f b;
        b[0] = sXn[ka * BB + b0 + lid];
        b[1] = sXn[(ka + 1) * BB + b0 + lid];
        c = __builtin_amdgcn_wmma_f32_16x16x4_f32(
                /*neg_a=*/false, a, /*neg_b=*/false, b,
                /*c_mod=*/(short)0, c, /*reuse_a=*/false, /*reuse_b=*/false);
    }

#pragma unroll
    for (int r = 0; r < 8; ++r)
        accbase[(o0 + r + half * 8) * BB + b0 + lid] = c[r];
}

__global__ void sigmoid_kernel(float* __restrict__ p, int n) {
    int i = blockIdx.x * blockDim.x + threadIdx.x;
    if (i < n) p[i] = 1.0f / (1.0f + __expf(-p[i]));
}

__global__ void tanh_kernel(float* __restrict__ p, int n) {
    int i = blockIdx.x * blockDim.x + threadIdx.x;
    if (i < n) p[i] = tanhf(p[i]);
}

// out[b, n*LAT + o] = -theta[n, o, b] * c3[n, o, b]
__global__ void final_kernel(const float* __restrict__ theta,
                             const float* __restrict__ c3,
                             float* __restrict__ out) {
    int idx = blockIdx.x * blockDim.x + threadIdx.x;   // over B*N*LAT
    if (idx >= BB * NN * LAT) return;
    int o = idx & 15;
    int n = (idx >> 4) & (NN - 1);
    int b = idx >> 16;                                 // idx / (N*LAT)
    size_t src = (size_t)n * (LAT * BB) + o * BB + b;
    out[idx] = -theta[src] * c3[src];
}

// ---------------------------------------------------------------------------
extern "C" void kernel_launch(void* const* d_in, const int* in_sizes, int n_in,
                              void* d_out, int out_size, void* d_ws, size_t ws_size,
                              hipStream_t stream) {
    // input order per setup_inputs():
    // 0 t_local, 1 y, 2 W_theta, 3 b_lat, 4 W_hid, 5 b_hid, 6 W_out,
    // 7 s1_rows, 8 s1_cols, 9 s1_vals, 10 s2_rows, 11 s2_cols, 12 s2_vals
    const float* y       = (const float*)d_in[1];
    const float* W_theta = (const float*)d_in[2];
    const float* b_lat   = (const float*)d_in[3];
    const float* W_hid   = (const float*)d_in[4];
    const float* b_hid   = (const float*)d_in[5];
    const float* W_out   = (const float*)d_in[6];
    const int*   s1r = (const int*)d_in[7];
    const int*   s1c = (const int*)d_in[8];
    const float* s1v = (const float*)d_in[9];
    const int*   s2r = (const int*)d_in[10];
    const int*   s2c = (const int*)d_in[11];
    const float* s2v = (const float*)d_in[12];
    float* out = (float*)d_out;

    // workspace layout (floats)
    const size_t SZ_LAT  = (size_t)NN * LAT * BB;     //  2,097,152
    const size_t SZ_UNIT = (size_t)NN * UNITS * BB;   //  8,388,608
    float* ws     = (float*)d_ws;
    float* x0_lat = ws;                     // (N, 16*32)
    float* t1     = x0_lat + SZ_LAT;        // (N, 64*32) max-size diffusion temp
    float* t2     = t1 + SZ_UNIT;           // (N, 64*32)
    float* theta  = t2 + SZ_UNIT;           // (N, 16, 32)
    float* c_unit = theta + SZ_LAT;         // (N, 64, 32) gconv2 out / gconv3 in
    float* c3     = c_unit + SZ_UNIT;       // (N, 16, 32)

    const int T = 256;
    auto blocks = [](size_t n, int t) { return (int)((n + t - 1) / t); };

    auto spmm = [&](const int* r, const int* c, const float* v,
                    const float* x, float* o, int FB, float scale) {
        size_t th = (size_t)NNZE * (FB >> 2);
        spmm_atomic_kernel<<<blocks(th, T), T, 0, stream>>>(r, c, v, x, o, FB, scale);
    };
    auto zero = [&](float* p, size_t n) {
        fill_zero_kernel<<<blocks(n >> 2, T), T, 0, stream>>>((float4*)p, (int)(n >> 2));
    };
    auto negc = [&](float* d, const float* s, size_t n) {
        neg_copy_kernel<<<blocks(n >> 2, T), T, 0, stream>>>((float4*)d, (const float4*)s,
                                                             (int)(n >> 2));
    };
    // GEMM template dispatch: (F, OUT) in {(16,16), (16,64), (64,16)}
    auto gemm16_16 = [&](const float* X, const float* W, float* acc, int m) {
        gemm_wmma_lds_kernel<16, 16><<<NN / 4, T, 0, stream>>>(X, W, acc, m);
    };
    auto gemm16_64 = [&](const float* X, const float* W, float* acc, int m) {
        gemm_wmma_lds_kernel<16, 64><<<NN / 1, T, 0, stream>>>(X, W, acc, m);
    };
    auto gemm64_16 = [&](const float* X, const float* W, float* acc, int m) {
        gemm_wmma_lds_kernel<64, 16><<<NN / 4, T, 0, stream>>>(X, W, acc, m);
    };

    // ---- x0 in (N, F, B) layout --------------------------------------------
    permute_y_kernel<<<blocks(SZ_LAT, T), T, 0, stream>>>(y, x0_lat);

    // ---- gconv1 (theta) & gconv2 (c_unit) share diffusion terms ------------
    bias_init_kernel<<<blocks(SZ_LAT, T), T, 0, stream>>>(theta, b_lat, LAT);
    bias_init_kernel<<<blocks(SZ_UNIT, T), T, 0, stream>>>(c_unit, b_hid, UNITS);

    // m = 0 : x0
    gemm16_16(x0_lat, W_theta, theta, 0);
    gemm16_64(x0_lat, W_hid, c_unit, 0);
    // support 1: x1 = S1 x0
    zero(t1, SZ_LAT);
    spmm(s1r, s1c, s1v, x0_lat, t1, LAT * BB, 1.0f);
    gemm16_16(t1, W_theta, theta, 1);
    gemm16_64(t1, W_hid, c_unit, 1);
    // x2 = 2 S1 x1 - x0
    negc(t2, x0_lat, SZ_LAT);
    spmm(s1r, s1c, s1v, t1, t2, LAT * BB, 2.0f);
    gemm16_16(t2, W_theta, theta, 2);
    gemm16_64(t2, W_hid, c_unit, 2);
    // support 2
    zero(t1, SZ_LAT);
    spmm(s2r, s2c, s2v, x0_lat, t1, LAT * BB, 1.0f);
    gemm16_16(t1, W_theta, theta, 3);
    gemm16_64(t1, W_hid, c_unit, 3);
    negc(t2, x0_lat, SZ_LAT);
    spmm(s2r, s2c, s2v, t1, t2, LAT * BB, 2.0f);
    gemm16_16(t2, W_theta, theta, 4);
    gemm16_64(t2, W_hid, c_unit, 4);

    sigmoid_kernel<<<blocks(SZ_LAT, T), T, 0, stream>>>(theta, (int)SZ_LAT);
    tanh_kernel<<<blocks(SZ_UNIT, T), T, 0, stream>>>(c_unit, (int)SZ_UNIT);

    // ---- gconv3: input c_unit (N, 64, 32), weights W_out (320,16), bias b_lat
    bias_init_kernel<<<blocks(SZ_LAT, T), T, 0, stream>>>(c3, b_lat, LAT);
    gemm64_16(c_unit, W_out, c3, 0);
    zero(t1, SZ_UNIT);
    spmm(s1r, s1c, s1v, c_unit, t1, UNITS * BB, 1.0f);
    gemm64_16(t1, W_out, c3, 1);
    negc(t2, c_unit, SZ_UNIT);
    spmm(s1r, s1c, s1v, t1, t2, UNITS * BB, 2.0f);
    gemm64_16(t2, W_out, c3, 2);
    zero(t1, SZ_UNIT);
    spmm(s2r, s2c, s2v, c_unit, t1, UNITS * BB, 1.0f);
    gemm64_16(t1, W_out, c3, 3);
    negc(t2, c_unit, SZ_UNIT);
    spmm(s2r, s2c, s2v, t1, t2, UNITS * BB, 2.0f);
    gemm64_16(t2, W_out, c3, 4);
    tanh_kernel<<<blocks(SZ_LAT, T), T, 0, stream>>>(c3, (int)SZ_LAT);

    // ---- out = -theta * c3, back to (B, N*LAT) -----------------------------
    final_kernel<<<blocks((size_t)out_size, T), T, 0, stream>>>(theta, c3, out);
}